// UDCTModule_20572893348233
// MI455X (gfx1250) — compile-verified
//
#include <hip/hip_runtime.h>
#include <hip/hip_bf16.h>

// ---------------------------------------------------------------------------
// UDCT forward transform for MI455X (gfx1250, wave32).
//   out[b] = Re( IFFT2( W_b .* FFT2(image) ) ),  b = 0..42, N = 512
// 512-point line FFT factored 512 = 2*16*16; the two DFT-16 stages run as
// complex 16x16 x 16x32 matmuls on V_WMMA_F32_16X16X4_F32 (fp32 matrix pipe).
// ---------------------------------------------------------------------------

#define FN   512
#define NB   43
#define LPB  4          // lines (FFT columns/rows) per 128-thread block
#define TPB  128
#define PI_F 3.14159265358979323846f

typedef float v2f __attribute__((ext_vector_type(2)));
typedef float v8f __attribute__((ext_vector_type(8)));

__device__ __forceinline__ v8f wmma_f32(v2f a, v2f b, v8f c) {
  // D = A(16x4) * B(4x16) + C(16x16), fp32, wave32
  return __builtin_amdgcn_wmma_f32_16x16x4_f32(false, a, false, b,
                                               (short)0, c, false, false);
}

// DFT-16 matrix held as per-lane WMMA A-fragments (4 K-chunks of K=4).
// A layout (ISA 7.12.2): lanes 0-15 -> M=lane, VGPR0=K0,VGPR1=K1;
//                        lanes16-31 -> M=lane-16, VGPR0=K2,VGPR1=K3.
struct W16 { v2f ar[4]; v2f ai[4]; v2f ain[4]; };

__device__ __forceinline__ void make_w16(float sgn, int lane, W16& w) {
  const int m  = lane & 15;
  const int kb = (lane >> 4) << 1;
  const float step = sgn * (2.0f * PI_F / 16.0f);
#pragma unroll
  for (int kc = 0; kc < 4; ++kc) {
    const int k0 = 4 * kc + kb;
    float s0, c0, s1, c1;
    sincosf(step * (float)(m * k0),       &s0, &c0);
    sincosf(step * (float)(m * (k0 + 1)), &s1, &c1);
    v2f tr; tr.x = c0;  tr.y = c1;  w.ar[kc]  = tr;
    v2f ti; ti.x = s0;  ti.y = s1;  w.ai[kc]  = ti;
    v2f tn; tn.x = -s0; tn.y = -s1; w.ain[kc] = tn;   // f32 WMMA has no A-neg
  }
}

// One 16x16 output tile of the complex matmul  Y = W16 * U,
// U in LDS with layout src[k*32 + col]; tile t covers cols 16t..16t+15.
// B layout mirrors A: lanes 0-15 N=lane K={0,1}; lanes 16-31 N=lane-16 K={2,3}.
__device__ __forceinline__ void ctile(const float* sr, const float* si, int t,
                                      int lane, const W16& w, v8f& cr, v8f& ci) {
  const int n  = 16 * t + (lane & 15);
  const int kb = (lane >> 4) << 1;
  v8f zr = {0.f,0.f,0.f,0.f,0.f,0.f,0.f,0.f};
  cr = zr; ci = zr;
#pragma unroll
  for (int kc = 0; kc < 4; ++kc) {
    const int k0 = 4 * kc + kb;
    v2f br; br.x = sr[k0 * 32 + n]; br.y = sr[(k0 + 1) * 32 + n];
    v2f bi; bi.x = si[k0 * 32 + n]; bi.y = si[(k0 + 1) * 32 + n];
    cr = wmma_f32(w.ar[kc],  br, cr);   // Yr += Wr*Ur
    cr = wmma_f32(w.ain[kc], bi, cr);   // Yr -= Wi*Ui
    ci = wmma_f32(w.ar[kc],  bi, ci);   // Yi += Wr*Ui
    ci = wmma_f32(w.ai[kc],  br, ci);   // Yi += Wi*Ur
  }
}

// Stage A: x[n1 + 16*m1 + 256*m2] -> radix-2 over m2, * w32^(m1*j2),
// store U[m1*32 + (n1 + 16*j2)]   (K-major layout for the stage-B matmul).
__device__ __forceinline__ void fft_stageA(const float* xr, const float* xi,
                                           float* ur, float* ui,
                                           int lane, float sgn) {
  const float tws = sgn * (2.0f * PI_F / 32.0f);
#pragma unroll
  for (int i = 0; i < 8; ++i) {
    const int p  = lane + 32 * i;           // [0,256)
    const int n1 = p & 15, m1 = p >> 4;
    const float a_r = xr[p],       a_i = xi[p];
    const float b_r = xr[p + 256], b_i = xi[p + 256];
    ur[m1 * 32 + n1] = a_r + b_r;
    ui[m1 * 32 + n1] = a_i + b_i;
    const float dr = a_r - b_r, di = a_i - b_i;
    float s, c; sincosf(tws * (float)m1, &s, &c);
    ur[m1 * 32 + n1 + 16] = dr * c - di * s;
    ui[m1 * 32 + n1 + 16] = dr * s + di * c;
  }
}

// Stage B (DFT16 over m1 via WMMA) + fused stage C (w512^(n1*k2) twiddle);
// writes V[n1*32 + k2] with k2 = 2*j1 + j2.
__device__ __forceinline__ void fft_stageBC(const float* ur, const float* ui,
                                            float* vr, float* vi,
                                            const float* twr, const float* twi,
                                            int lane, const W16& w) {
  const int nn = lane & 15;
  const int mb = (lane >> 4) * 8;           // C/D row base per ISA layout
#pragma unroll
  for (int t = 0; t < 2; ++t) {
    v8f gr, gi;
    ctile(ur, ui, t, lane, w, gr, gi);
    const int ccol = 16 * t + nn;
    const int n1 = ccol & 15, j2 = ccol >> 4;
#pragma unroll
    for (int r = 0; r < 8; ++r) {
      const int j1  = mb + r;
      const int k2  = 2 * j1 + j2;
      const int idx = n1 * 32 + k2;
      const float tr = twr[idx], ti_ = twi[idx];
      const float g_r = gr[r], g_i = gi[r];
      vr[idx] = g_r * tr - g_i * ti_;
      vi[idx] = g_r * ti_ + g_i * tr;
    }
  }
}

__device__ __forceinline__ void fill_tw512(float* twr, float* twi,
                                           int tid, float sgn) {
  const float step = sgn * (2.0f * PI_F / 512.0f);
  for (int i = tid; i < FN; i += TPB) {
    const int n1 = i >> 5, k2 = i & 31;
    float s, c; sincosf(step * (float)(n1 * k2), &s, &c);
    twr[i] = c; twi[i] = s;
  }
}

#define FFT_SHARED                                              \
  __shared__ float Ar[LPB][FN], Ai[LPB][FN];                    \
  __shared__ float Br[LPB][FN], Bi[LPB][FN];                    \
  __shared__ float Twr[FN], Twi[FN];

// ---------------- kernel 1: forward FFT over rows (real input) -------------
__global__ __launch_bounds__(TPB)
void k_fwd_rows(const float* __restrict__ img, float2* __restrict__ specRow) {
  FFT_SHARED
  const int tid = threadIdx.x, lane = tid & 31, wid = tid >> 5;
  const int line = blockIdx.x * LPB + wid;          // row index
  fill_tw512(Twr, Twi, tid, -1.0f);
  W16 w; make_w16(-1.0f, lane, w);
#pragma unroll
  for (int i = 0; i < 16; ++i) {                    // coalesced row load
    const int p = lane + 32 * i;
    Ar[wid][p] = img[line * FN + p];
    Ai[wid][p] = 0.0f;
  }
  __syncthreads();
  fft_stageA(Ar[wid], Ai[wid], Br[wid], Bi[wid], lane, -1.0f);
  __syncthreads();
  fft_stageBC(Br[wid], Bi[wid], Ar[wid], Ai[wid], Twr, Twi, lane, w);
  __syncthreads();
  const int nn = lane & 15, mb = (lane >> 4) * 8;
#pragma unroll
  for (int t = 0; t < 2; ++t) {
    v8f dr, di;
    ctile(Ar[wid], Ai[wid], t, lane, w, dr, di);
#pragma unroll
    for (int r = 0; r < 8; ++r) {
      const int o = 32 * (mb + r) + 16 * t + nn;
      specRow[line * FN + o] = make_float2(dr[r], di[r]);
    }
  }
}

// ---------------- kernel 2: forward FFT over columns -----------------------
__global__ __launch_bounds__(TPB)
void k_fwd_cols(const float2* __restrict__ specRow, float2* __restrict__ spec) {
  FFT_SHARED
  const int tid = threadIdx.x, lane = tid & 31, wid = tid >> 5;
  const int c0 = blockIdx.x * LPB;
  fill_tw512(Twr, Twi, tid, -1.0f);
  W16 w; make_w16(-1.0f, lane, w);
  for (int i = tid; i < FN * LPB; i += TPB) {       // cooperative tile load
    const int e = i >> 2, ww = i & 3;
    const float2 s = specRow[e * FN + c0 + ww];
    Ar[ww][e] = s.x; Ai[ww][e] = s.y;
  }
  __syncthreads();
  fft_stageA(Ar[wid], Ai[wid], Br[wid], Bi[wid], lane, -1.0f);
  __syncthreads();
  fft_stageBC(Br[wid], Bi[wid], Ar[wid], Ai[wid], Twr, Twi, lane, w);
  __syncthreads();
  const int nn = lane & 15, mb = (lane >> 4) * 8;
#pragma unroll
  for (int t = 0; t < 2; ++t) {
    v8f dr, di;
    ctile(Ar[wid], Ai[wid], t, lane, w, dr, di);
#pragma unroll
    for (int r = 0; r < 8; ++r) {
      const int o = 32 * (mb + r) + 16 * t + nn;
      Br[wid][o] = dr[r]; Bi[wid][o] = di[r];
    }
  }
  __syncthreads();
  for (int i = tid; i < FN * LPB; i += TPB) {       // cooperative tile store
    const int o = i >> 2, ww = i & 3;
    spec[o * FN + c0 + ww] = make_float2(Br[ww][o], Bi[ww][o]);
  }
}

// ------ kernel 3: per-band window multiply + inverse FFT over columns ------
__global__ __launch_bounds__(TPB)
void k_band_cols(const float2* __restrict__ spec, const float* __restrict__ win,
                 float2* __restrict__ T, int b0) {
  FFT_SHARED
  const int tid = threadIdx.x, lane = tid & 31, wid = tid >> 5;
  const int c0 = blockIdx.x * LPB;
  const int bi = blockIdx.y;
  const size_t woff = (size_t)(b0 + bi) * FN * FN;
  float2* Tb = T + (size_t)bi * FN * FN;
  fill_tw512(Twr, Twi, tid, 1.0f);
  W16 w; make_w16(1.0f, lane, w);
  for (int i = tid; i < FN * LPB; i += TPB) {       // fused window multiply
    const int e = i >> 2, ww = i & 3;
    const float2 s = spec[e * FN + c0 + ww];
    const float  wv = win[woff + (size_t)e * FN + c0 + ww];
    Ar[ww][e] = s.x * wv; Ai[ww][e] = s.y * wv;
  }
  __syncthreads();
  fft_stageA(Ar[wid], Ai[wid], Br[wid], Bi[wid], lane, 1.0f);
  __syncthreads();
  fft_stageBC(Br[wid], Bi[wid], Ar[wid], Ai[wid], Twr, Twi, lane, w);
  __syncthreads();
  const int nn = lane & 15, mb = (lane >> 4) * 8;
#pragma unroll
  for (int t = 0; t < 2; ++t) {
    v8f dr, di;
    ctile(Ar[wid], Ai[wid], t, lane, w, dr, di);
#pragma unroll
    for (int r = 0; r < 8; ++r) {
      const int o = 32 * (mb + r) + 16 * t + nn;
      Br[wid][o] = dr[r]; Bi[wid][o] = di[r];
    }
  }
  __syncthreads();
  for (int i = tid; i < FN * LPB; i += TPB) {
    const int o = i >> 2, ww = i & 3;
    Tb[o * FN + c0 + ww] = make_float2(Br[ww][o], Bi[ww][o]);
  }
}

// ------ kernel 4: inverse FFT over rows, real part * 1/N^2 -> output -------
__global__ __launch_bounds__(TPB)
void k_band_rows(const float2* __restrict__ T, float* __restrict__ out, int b0) {
  FFT_SHARED
  const int tid = threadIdx.x, lane = tid & 31, wid = tid >> 5;
  const int bi = blockIdx.y;
  const float2* Tb = T + (size_t)bi * FN * FN;
  float* ob = out + (size_t)(b0 + bi) * FN * FN;
  const int line = blockIdx.x * LPB + wid;          // row index
  fill_tw512(Twr, Twi, tid, 1.0f);
  W16 w; make_w16(1.0f, lane, w);
#pragma unroll
  for (int i = 0; i < 16; ++i) {                    // coalesced row load
    const int p = lane + 32 * i;
    const float2 v = Tb[line * FN + p];
    Ar[wid][p] = v.x; Ai[wid][p] = v.y;
  }
  __syncthreads();
  fft_stageA(Ar[wid], Ai[wid], Br[wid], Bi[wid], lane, 1.0f);
  __syncthreads();
  fft_stageBC(Br[wid], Bi[wid], Ar[wid], Ai[wid], Twr, Twi, lane, w);
  __syncthreads();
  const float scale = 1.0f / (512.0f * 512.0f);
  const int nn = lane & 15, mb = (lane >> 4) * 8;
#pragma unroll
  for (int t = 0; t < 2; ++t) {
    v8f dr, di;
    ctile(Ar[wid], Ai[wid], t, lane, w, dr, di);
#pragma unroll
    for (int r = 0; r < 8; ++r) {
      const int o = 32 * (mb + r) + 16 * t + nn;
      ob[line * FN + o] = dr[r] * scale;            // Hermitian => real coeffs
    }
  }
}

// ---------------------------------------------------------------------------
extern "C" void kernel_launch(void* const* d_in, const int* in_sizes, int n_in,
                              void* d_out, int out_size, void* d_ws,
                              size_t ws_size, hipStream_t stream) {
  (void)in_sizes; (void)n_in; (void)out_size;
  const float* img = (const float*)d_in[0];   // [512,512] f32
  const float* win = (const float*)d_in[1];   // [43,512,512] f32
  float* out = (float*)d_out;                 // [43*512*512] f32

  // Workspace: specRow (2MB) | spec (2MB) | band intermediates (2MB each)
  float2* specRow = (float2*)d_ws;
  float2* spec    = specRow + (size_t)FN * FN;
  float2* bandT   = spec    + (size_t)FN * FN;
  const size_t reserved = 2ull * sizeof(float2) * FN * FN;
  const size_t perBand  = sizeof(float2) * (size_t)FN * FN;
  int chunk = (ws_size > reserved) ? (int)((ws_size - reserved) / perBand) : 0;
  if (chunk < 1)  chunk = 1;
  if (chunk > NB) chunk = NB;

  const dim3 blk(TPB);
  k_fwd_rows<<<dim3(FN / LPB), blk, 0, stream>>>(img, specRow);
  k_fwd_cols<<<dim3(FN / LPB), blk, 0, stream>>>(specRow, spec);
  for (int b0 = 0; b0 < NB; b0 += chunk) {
    const int nb = (NB - b0 < chunk) ? (NB - b0) : chunk;
    k_band_cols<<<dim3(FN / LPB, nb), blk, 0, stream>>>(spec, win, bandT, b0);
    k_band_rows<<<dim3(FN / LPB, nb), blk, 0, stream>>>(bandT, out, b0);
  }
}